// Model_53403623358837
// MI455X (gfx1250) — compile-verified
//
#include <hip/hip_runtime.h>
#include <hip/hip_bf16.h>

// ---------------------------------------------------------------------------
// Model: per-feature Conv1d(1,16,8)+ReLU -> reshape -> 4x dense+ReLU -> 2 proj.
// MI455X roofline: ~1.75 GB of f32 weights streamed exactly once ==> ~75us
// HBM-bound floor @ 23.3 TB/s; total ~28 GFLOP is negligible vs the matrix
// ceiling.  Therefore: stay FP32 (no conversion traffic), use
// v_wmma_f32_16x16x4_f32 on the matrix pipe, and hide the activation staging
// behind compute with gfx1250 async-to-LDS copies (ASYNCcnt double buffering).
// ---------------------------------------------------------------------------

typedef float v2f __attribute__((ext_vector_type(2)));
typedef float v8f __attribute__((ext_vector_type(8)));

#define Bsz   32
#define Ssz   343
#define Esz   5
#define Fsz   4
#define Ksz   8
#define Csz   16
#define Lsz   336
#define FLAT  21504   // C*F*L
#define H1    16384
#define H2    4096
#define H4    64
#define LBL   16

// ---------------------------------------------------------------------------
// Stage 0: conv + ReLU + scatter into the reference's reshape layout.
//   X[row, col], row = 2*c + (b>=16), col = ((b&15)*4 + f)*336 + l
// ---------------------------------------------------------------------------
__global__ void conv_scatter_kernel(const float* __restrict__ in,
                                    const float* __restrict__ cw,
                                    const float* __restrict__ cb,
                                    float* __restrict__ X) {
    int tid = blockIdx.x * blockDim.x + threadIdx.x;
    if (tid >= Csz * Bsz * Fsz * Lsz) return;
    int l = tid % Lsz;
    int f = (tid / Lsz) & (Fsz - 1);
    int b = (tid / (Lsz * Fsz)) & (Bsz - 1);
    int c = tid / (Lsz * Fsz * Bsz);

    float s = cb[c];
#pragma unroll
    for (int t = 0; t < Ksz; ++t)
        s = fmaf(in[b * (Ssz * Esz) + (l + t) * Esz + f], cw[c * Ksz + t], s);
    s = fmaxf(s, 0.0f);

    int row = 2 * c + (b >> 4);
    int col = ((b & 15) * Fsz + f) * Lsz + l;
    X[row * FLAT + col] = s;
}

// ---------------------------------------------------------------------------
// Skinny GEMM:  Y[32, N] = act( X[32, K] @ W[N, K]^T + bias )
//  - Each wave owns a 16-wide N tile; M=32 -> two accumulators share one
//    streamed B fragment (weights read exactly once from HBM).
//  - X K-chunks double-buffered in LDS via global_load_async_to_lds_b128
//    (ASYNCcnt); one barrier per chunk, copy overlapped with WMMA compute.
//  - +4 float row padding => A-fragment ds reads hit 64 distinct banks.
//  - v_wmma_f32_16x16x4_f32 fragment ownership per CDNA5 ISA 7.12.2:
//      A: lane m = l&15, VGPR j holds K = 2*(l>>4)+j
//      B: lane n = l&15, VGPR j holds K = 2*(l>>4)+j (mirrored)
//      D: lane n = l&15, VGPR r holds row M = 8*(l>>4)+r
// ---------------------------------------------------------------------------
template <int KCHUNK, bool RELU>
__global__ __launch_bounds__(128) void gemm_wmma_f32(
    const float* __restrict__ X,     // [32, K] row-major
    const float* __restrict__ W,     // [N, K]  row-major (torch Linear)
    const float* __restrict__ bias,  // [N]
    float* __restrict__ Y,           // [32, N] row-major
    int K, int N) {
    constexpr int PAD = 4;
    constexpr int RS  = KCHUNK + PAD;        // padded row stride (floats)
    __shared__ float Xs[2][32 * RS];

    const int lane  = threadIdx.x & 31;
    const int wave  = threadIdx.x >> 5;
    const int nwav  = blockDim.x >> 5;
    const int n     = (blockIdx.x * nwav + wave) * 16 + (lane & 15);
    const int khalf = (lane >> 4) * 2;       // 0 or 2
    const int m     = lane & 15;

    v8f acc0 = {};   // rows 0..15
    v8f acc1 = {};   // rows 16..31

    // Async-stage one 32xKCHUNK chunk of X into LDS buffer `buf`.
    // LDS dest addr = low 32 bits of the generic shared pointer;
    // global source in a 64-bit VGPR pair.
    auto stage = [&](int k0, int buf) {
        for (int idx = threadIdx.x * 4; idx < 32 * KCHUNK; idx += blockDim.x * 4) {
            int r = idx / KCHUNK, c = idx % KCHUNK;
            unsigned     lds = (unsigned)(uintptr_t)(&Xs[buf][r * RS + c]);
            const float* src = X + r * K + k0 + c;
            asm volatile("global_load_async_to_lds_b128 %0, %1, off"
                         :: "v"(lds), "v"(src) : "memory");
        }
    };

    stage(0, 0);
    const int nchunks = K / KCHUNK;
    for (int ci = 0; ci < nchunks; ++ci) {
        const int k0  = ci * KCHUNK;
        const int buf = ci & 1;

        asm volatile("s_wait_asynccnt 0x0" ::: "memory");
        __syncthreads();                      // buf ready; other buf free
        if (ci + 1 < nchunks) stage(k0 + KCHUNK, buf ^ 1);

        const float* wrow = W + (size_t)n * K + k0 + khalf;
        if (k0 + KCHUNK < K)                  // pull next weight lines
            __builtin_prefetch(wrow + KCHUNK, 0, 1);   // global_prefetch_b8

        const float* xs = &Xs[buf][0];
#pragma unroll 4
        for (int kk = 0; kk < KCHUNK; kk += 4) {
            v2f a0 = *(const v2f*)(xs + m * RS + kk + khalf);
            v2f a1 = *(const v2f*)(xs + (16 + m) * RS + kk + khalf);
            v2f bf = *(const v2f*)(wrow + kk);
            acc0 = __builtin_amdgcn_wmma_f32_16x16x4_f32(
                false, a0, false, bf, (short)0, acc0, false, false);
            acc1 = __builtin_amdgcn_wmma_f32_16x16x4_f32(
                false, a1, false, bf, (short)0, acc1, false, false);
        }
        // no trailing barrier: next iteration's wait+barrier protects buffers
    }

    const float bv    = bias[n];
    const int   mbase = (lane >> 4) * 8;
#pragma unroll
    for (int r = 0; r < 8; ++r) {
        float y0 = acc0[r] + bv;
        float y1 = acc1[r] + bv;
        if (RELU) { y0 = fmaxf(y0, 0.0f); y1 = fmaxf(y1, 0.0f); }
        Y[(mbase + r) * N + n]      = y0;   // rows 0..15
        Y[(16 + mbase + r) * N + n] = y1;   // rows 16..31
    }
}

// ---------------------------------------------------------------------------
// Final 16 -> 1 projection: out[m] = Y5[m,:] . wf + bf   (32 threads)
// ---------------------------------------------------------------------------
__global__ void final_wf_kernel(const float* __restrict__ Y5,
                                const float* __restrict__ wf,
                                const float* __restrict__ bf,
                                float* __restrict__ out) {
    int mrow = threadIdx.x;
    if (mrow >= Bsz) return;
    float s = bf[0];
#pragma unroll
    for (int nn = 0; nn < LBL; ++nn)
        s = fmaf(Y5[mrow * LBL + nn], wf[nn], s);
    out[mrow] = s;
}

extern "C" void kernel_launch(void* const* d_in, const int* in_sizes, int n_in,
                              void* d_out, int out_size, void* d_ws, size_t ws_size,
                              hipStream_t stream) {
    const float* inputs = (const float*)d_in[0];
    const float* conv_w = (const float*)d_in[4];
    const float* conv_b = (const float*)d_in[5];
    const float* w1 = (const float*)d_in[6];
    const float* b1 = (const float*)d_in[7];
    const float* w2 = (const float*)d_in[8];
    const float* b2 = (const float*)d_in[9];
    const float* w3 = (const float*)d_in[10];
    const float* b3 = (const float*)d_in[11];
    const float* w4 = (const float*)d_in[12];
    const float* b4 = (const float*)d_in[13];
    const float* wo = (const float*)d_in[14];
    const float* bo = (const float*)d_in[15];
    const float* wf = (const float*)d_in[16];
    const float* bf = (const float*)d_in[17];
    float* out = (float*)d_out;

    // Workspace layout (floats)
    float* X  = (float*)d_ws;           // 32*21504
    float* Y1 = X  + Bsz * FLAT;        // 32*16384
    float* Y2 = Y1 + Bsz * H1;          // 32*4096
    float* Y3 = Y2 + Bsz * H2;          // 32*4096
    float* Y4 = Y3 + Bsz * H2;          // 32*64
    float* Y5 = Y4 + Bsz * H4;          // 32*16

    // Stage 0: conv + reshape  (16*32*4*336 = 688128 threads)
    conv_scatter_kernel<<<(Csz * Bsz * Fsz * Lsz) / 256, 256, 0, stream>>>(
        inputs, conv_w, conv_b, X);

    // dense1: 32x21504 @ 21504x16384  (1.41 GB weights -> dominant, HBM-bound)
    gemm_wmma_f32<128, true><<<H1 / 64, 128, 0, stream>>>(X, w1, b1, Y1, FLAT, H1);
    // dense2: 32x16384 @ 16384x4096
    gemm_wmma_f32<128, true><<<H2 / 64, 128, 0, stream>>>(Y1, w2, b2, Y2, H1, H2);
    // dense3: 32x4096 @ 4096x4096
    gemm_wmma_f32<128, true><<<H2 / 64, 128, 0, stream>>>(Y2, w3, b3, Y3, H2, H2);
    // dense4: 32x4096 @ 4096x64  (one block, 4 waves cover N=64)
    gemm_wmma_f32<128, true><<<1, 128, 0, stream>>>(Y3, w4, b4, Y4, H2, H4);
    // output_layer: 32x64 @ 64x16  (single wave, EXEC all-ones for WMMA)
    gemm_wmma_f32<64, false><<<1, 32, 0, stream>>>(Y4, wo, bo, Y5, H4, LBL);
    // output_layerf: 32x16 @ 16x1
    final_wf_kernel<<<1, 32, 0, stream>>>(Y5, wf, bf, out);
}